// HistoLoss_63806034149742
// MI455X (gfx1250) — compile-verified
//
#include <hip/hip_runtime.h>
#include <cstdint>
#include <cstddef>

// Problem constants (from the reference)
#define N_DIM   2048
#define T_DIM   256
#define D_DIM   64
#define NB      64
#define BLK     512                     // 16 wave32 per block
#define ROWS_PER_TILE (BLK / 16)        // 32 rows * 64 cols = 2048 floats/tile
#define NT      (N_DIM / ROWS_PER_TILE) // 64 tiles per block

// Async global->LDS staging (CDNA5 GLOBAL_LOAD_ASYNC_TO_LDS_B128, ASYNCcnt).
// lds_off is the wave-relative LDS byte offset (low 32 bits of a generic
// pointer into __shared__), g is the 64-bit global address.
__device__ __forceinline__ void async_load_b128(unsigned lds_off, const float* g) {
    unsigned long long ga = (unsigned long long)(uintptr_t)g;
    asm volatile("global_load_async_to_lds_b128 %0, %1, off"
                 :: "v"(lds_off), "v"(ga)
                 : "memory");
}

__device__ __forceinline__ void wait_async_le1() {
    asm volatile("s_wait_asynccnt 1" ::: "memory");
}
__device__ __forceinline__ void wait_async_le0() {
    asm volatile("s_wait_asynccnt 0" ::: "memory");
}

// One block per t. Builds the full [D=64][NB=64] histogram for that t in LDS
// (ds_add_u32, privatized, deterministic), then fuses the |hist/N - dens|
// partial reduction for that t. No global atomics anywhere.
__global__ __launch_bounds__(BLK) void histo_loss_block_kernel(
    const float* __restrict__ x,      // [N, T, D]
    const float* __restrict__ dens,   // [T, D, NB]
    const float* __restrict__ bmin,   // [T, D]
    const float* __restrict__ bmax,   // [T, D]
    float* __restrict__ partial)      // [T]
{
    __shared__ unsigned hist[D_DIM * NB];   // 16 KB
    __shared__ float    stage[2][BLK * 4];  // 2 x 8 KB double buffer
    __shared__ float    red[BLK];           // 2 KB

    const int tid       = threadIdx.x;
    const int t         = blockIdx.x;
    const int rowInTile = tid >> 4;         // 0..31
    const int c0        = (tid & 15) * 4;   // column group: 4 consecutive d

    // Zero private histogram
    #pragma unroll
    for (int k = 0; k < (D_DIM * NB) / BLK; ++k)
        hist[tid + k * BLK] = 0u;

    // Per-thread bin parameters for its 4 columns (loop-invariant over n)
    float av[4], bv[4], sc[4];
    #pragma unroll
    for (int k = 0; k < 4; ++k) {
        float a = bmin[t * D_DIM + c0 + k];
        float b = bmax[t * D_DIM + c0 + k];
        av[k] = a;
        bv[k] = b;
        sc[k] = 64.0f / (b - a);            // (x-a)*sc == (x-a)/w*n_bins (<=1ulp)
    }
    __syncthreads();

    // Async-staging pipeline. Each thread's b128 covers exactly the 4 floats
    // it will consume, so the per-wave s_wait_asynccnt is the only sync needed.
    const unsigned lds0 = (unsigned)(uintptr_t)&stage[0][tid * 4];
    const unsigned lds1 = (unsigned)(uintptr_t)&stage[1][tid * 4];
    const float* gbase = x + ((size_t)rowInTile * T_DIM + (size_t)t) * D_DIM + c0;
    const size_t tileStride = (size_t)ROWS_PER_TILE * T_DIM * D_DIM;

    async_load_b128(lds0, gbase);
    for (int i = 0; i < NT; ++i) {
        if (i + 1 < NT) {
            async_load_b128(((i + 1) & 1) ? lds1 : lds0,
                            gbase + (size_t)(i + 1) * tileStride);
            wait_async_le1();               // tile i complete (in-order loads)
        } else {
            wait_async_le0();
        }
        const float* s = &stage[i & 1][tid * 4];
        #pragma unroll
        for (int k = 0; k < 4; ++k) {
            float xv  = s[k];
            bool  inr = (xv >= av[k]) && (xv <= bv[k]);   // histc range rule
            float v   = floorf((xv - av[k]) * sc[k]);
            v = fminf(fmaxf(v, 0.0f), 63.0f);             // clip to [0,63]
            int bin = (int)v;
            if (inr) atomicAdd(&hist[(c0 + k) * NB + bin], 1u);  // ds_add_u32
        }
    }
    __syncthreads();

    // Fused per-t loss partial: sum over 4096 (d,bin) cells
    float local = 0.0f;
    const float invN = 1.0f / (float)N_DIM;
    const float* dt = dens + (size_t)t * (D_DIM * NB);
    #pragma unroll
    for (int k = 0; k < (D_DIM * NB) / BLK; ++k) {
        int idx = tid + k * BLK;
        float df = (float)hist[idx] * invN;
        local += fabsf(df - dt[idx]);
    }
    red[tid] = local;
    __syncthreads();
    #pragma unroll
    for (int sN = BLK / 2; sN > 0; sN >>= 1) {
        if (tid < sN) red[tid] += red[tid + sN];
        __syncthreads();
    }
    if (tid == 0) partial[t] = red[0];
}

// Single block: deterministic fixed-order sum of the 256 per-t partials.
__global__ __launch_bounds__(256) void histo_loss_finalize_kernel(
    const float* __restrict__ partial, float* __restrict__ out)
{
    __shared__ float red[256];
    const int tid = threadIdx.x;
    red[tid] = partial[tid];
    __syncthreads();
    #pragma unroll
    for (int sN = 128; sN > 0; sN >>= 1) {
        if (tid < sN) red[tid] += red[tid + sN];
        __syncthreads();
    }
    if (tid == 0)
        out[0] = red[0] * (1.0f / (float)(T_DIM * D_DIM * NB));  // REG = 1.0
}

extern "C" void kernel_launch(void* const* d_in, const int* in_sizes, int n_in,
                              void* d_out, int out_size, void* d_ws, size_t ws_size,
                              hipStream_t stream) {
    (void)in_sizes; (void)n_in; (void)out_size; (void)ws_size;
    const float* x    = (const float*)d_in[0];   // x_fake   [N,T,D]
    const float* dens = (const float*)d_in[1];   // densities[T,D,NB]
    const float* bmin = (const float*)d_in[2];   // bin_min  [T,D]
    const float* bmax = (const float*)d_in[3];   // bin_max  [T,D]
    float* out     = (float*)d_out;
    float* partial = (float*)d_ws;               // 256 floats of scratch

    histo_loss_block_kernel<<<T_DIM, BLK, 0, stream>>>(x, dens, bmin, bmax, partial);
    histo_loss_finalize_kernel<<<1, 256, 0, stream>>>(partial, out);
}